// MatMulRemapper_87840671137910
// MI455X (gfx1250) — compile-verified
//
#include <hip/hip_runtime.h>
#include <hip/hip_bf16.h>

typedef float v2f __attribute__((ext_vector_type(2)));
typedef float v8f __attribute__((ext_vector_type(8)));

#define N_   4
#define C_   128
#define H_   64
#define W_   64
#define HW_  (H_ * W_)
#define EPS_ 0.001f

// One block = one (n, 16-wide k tile). 256 threads = 8 wave32s; wave w owns
// channels [16w, 16w+16). Threads 0..15 precompute the 4 corner indices +
// weights for the 16 k positions into LDS (shared by all 8 waves / 128
// channels). Each wave then accumulates its 16x16 (channel x k) f32 output
// tile with 16 chained V_WMMA_F32_16X16X4_F32 ops: WMMA g consumes the 4
// gathered x values of position g (A, 16x4) and a one-hot B (4x16) whose
// column g carries the 4 weights.
__global__ __launch_bounds__(256)
void remap_wmma_kernel(const float* __restrict__ x,
                       const float* __restrict__ grid,
                       float* __restrict__ out) {
    __shared__ int   s_idx[16][4];
    __shared__ float s_w[16][4];

    const int tid    = threadIdx.x;
    const int n      = blockIdx.x / (HW_ / 16);
    const int k_base = (blockIdx.x % (HW_ / 16)) * 16;

    if (tid < 16) {
        const int j = k_base + tid;
        // validity requires in-range for ALL batches (matches .all(0))
        bool valid = true;
        float gx_n = 0.f, gy_n = 0.f;
        #pragma unroll
        for (int b = 0; b < N_; ++b) {
            const float gx = grid[((size_t)b * HW_ + j) * 2 + 0];
            const float gy = grid[((size_t)b * HW_ + j) * 2 + 1];
            valid = valid && (gx >= -EPS_) && (gx <= (float)(H_ - 1) + EPS_)
                          && (gy >= -EPS_) && (gy <= (float)(W_ - 1) + EPS_);
            if (b == n) { gx_n = gx; gy_n = gy; }
        }
        const float gxc = fminf(fmaxf(gx_n, EPS_), (float)(H_ - 1) - EPS_);
        const float gyc = fminf(fmaxf(gy_n, EPS_), (float)(W_ - 1) - EPS_);
        const float fxi = floorf(gxc);
        const float fyi = floorf(gyc);
        const int   ix  = (int)fxi;
        const int   iy  = (int)fyi;
        const float fx  = gxc - fxi;
        const float fy  = gyc - fyi;
        const int base  = ix * W_ + iy;
        s_idx[tid][0] = base;            // (0,0)
        s_idx[tid][1] = base + 1;        // (0,1)
        s_idx[tid][2] = base + W_;       // (1,0)
        s_idx[tid][3] = base + W_ + 1;   // (1,1)
        const float vm = valid ? 1.f : 0.f;
        // faithful to source: coeffs = outer([1-fx, fx], [fx, fy])
        s_w[tid][0] = (1.f - fx) * fx * vm;
        s_w[tid][1] = (1.f - fx) * fy * vm;
        s_w[tid][2] = fx * fx * vm;
        s_w[tid][3] = fx * fy * vm;
    }
    __syncthreads();

    const int lane  = tid & 31;
    const int wave  = tid >> 5;          // c-tile id, 0..7
    const int half  = lane >> 4;         // 0 -> K=0,1 ; 1 -> K=2,3
    const int laneM = lane & 15;         // A row (channel) / B-D column (k)
    const int c     = wave * 16 + laneM;

    const float* __restrict__ xrow = x + ((size_t)n * C_ + c) * (size_t)HW_;

    v8f acc = {0.f, 0.f, 0.f, 0.f, 0.f, 0.f, 0.f, 0.f};

    #pragma unroll
    for (int g = 0; g < 16; ++g) {
        const int   i0 = s_idx[g][half * 2 + 0];
        const int   i1 = s_idx[g][half * 2 + 1];
        const float a0 = xrow[i0];
        const float a1 = xrow[i1];
        const float sel = (laneM == g) ? 1.f : 0.f;
        const float b0 = s_w[g][half * 2 + 0] * sel;
        const float b1 = s_w[g][half * 2 + 1] * sel;
        v2f A = {a0, a1};
        v2f B = {b0, b1};
        // (neg_a, A, neg_b, B, c_mod, C, reuse_a, reuse_b)
        acc = __builtin_amdgcn_wmma_f32_16x16x4_f32(
            false, A, false, B, (short)0, acc, false, false);
    }

    // D 16x16 f32: lanes 0-15 column N=lane rows 0..7; lanes 16-31 column
    // N=lane-16 rows 8..15 (VGPR r -> M = r + 8*half).
    #pragma unroll
    for (int r = 0; r < 8; ++r) {
        const int m = r + 8 * half;                    // channel within tile
        out[((size_t)n * C_ + wave * 16 + m) * (size_t)HW_ + k_base + laneM] =
            acc[r];
    }
}

extern "C" void kernel_launch(void* const* d_in, const int* in_sizes, int n_in,
                              void* d_out, int out_size, void* d_ws, size_t ws_size,
                              hipStream_t stream) {
    (void)in_sizes; (void)n_in; (void)out_size; (void)d_ws; (void)ws_size;
    const float* x    = (const float*)d_in[0];   // (N, C, HW) f32
    const float* grid = (const float*)d_in[1];   // (N, H, W, 2) f32
    float*       out  = (float*)d_out;           // (N, C, HW) f32

    const int blocks = N_ * (HW_ / 16);          // 1024
    remap_wmma_kernel<<<blocks, 256, 0, stream>>>(x, grid, out);
}